// FraudGNN_4234837753868
// MI455X (gfx1250) — compile-verified
//
#include <hip/hip_runtime.h>
#include <math.h>

typedef __attribute__((ext_vector_type(2))) float v2f;
typedef __attribute__((ext_vector_type(8))) float v8f;

// ---------- helpers ----------
__device__ __forceinline__ unsigned fenc(float f) {
    unsigned u = __float_as_uint(f);
    return (u & 0x80000000u) ? ~u : (u | 0x80000000u);
}
__device__ __forceinline__ float fdec(unsigned u) {
    return __uint_as_float((u & 0x80000000u) ? (u & 0x7FFFFFFFu) : ~u);
}
__device__ __forceinline__ float lrelu(float x) { return x > 0.f ? x : 0.2f * x; }
__device__ __forceinline__ float elu(float x)   { return x > 0.f ? x : (__expf(x) - 1.f); }

// ---------- fills ----------
__global__ void fill_f32(float* p, float v, long n) {
    long i = (long)blockIdx.x * blockDim.x + threadIdx.x;
    if (i < n) p[i] = v;
}
__global__ void fill_u32(unsigned* p, unsigned v, long n) {
    long i = (long)blockIdx.x * blockDim.x + threadIdx.x;
    if (i < n) p[i] = v;
}

// ---------- fp32 WMMA GEMM: C[M,NC] = A[M,K] @ B[K,NC], one wave per 16x16 tile ----------
__global__ void wmma_gemm_f32(const float* __restrict__ A, const float* __restrict__ B,
                              float* __restrict__ C, int M, int K, int NC) {
    const int lane = threadIdx.x & 31;
    const int grp  = lane >> 4;    // 0: K={0,1}, 1: K={2,3}
    const int l    = lane & 15;    // M (for A) / N (for B) within tile
    const int m0   = blockIdx.x * 16;
    const int n0   = blockIdx.y * 16;

    int ra = m0 + l; if (ra >= M) ra = M - 1;  // clamp: keep EXEC all-ones for WMMA

    v8f acc = {0.f, 0.f, 0.f, 0.f, 0.f, 0.f, 0.f, 0.f};
    for (int k0 = 0; k0 < K; k0 += 4) {
        v2f a, b;
        a.x = A[(size_t)ra * K + k0 + 2 * grp];
        a.y = A[(size_t)ra * K + k0 + 2 * grp + 1];
        b.x = B[(size_t)(k0 + 2 * grp)     * NC + n0 + l];
        b.y = B[(size_t)(k0 + 2 * grp + 1) * NC + n0 + l];
        acc = __builtin_amdgcn_wmma_f32_16x16x4_f32(false, a, false, b,
                                                    (short)0, acc, false, false);
    }
#pragma unroll
    for (int v = 0; v < 8; ++v) {
        int row = m0 + v + 8 * grp;
        if (row < M) C[(size_t)row * NC + n0 + l] = acc[v];
    }
}

// ---------- per-node attention coefficients ----------
template <int H, int C>
__global__ void alpha_kernel(const float* __restrict__ xl, const float* __restrict__ a_src,
                             const float* __restrict__ a_dst, float* __restrict__ as,
                             float* __restrict__ ad, int N) {
    int idx = blockIdx.x * blockDim.x + threadIdx.x;
    if (idx >= N * H) return;
    int n = idx / H, h = idx % H;
    const float* row = xl + (size_t)n * H * C + (size_t)h * C;
    float s = 0.f, d = 0.f;
#pragma unroll
    for (int c = 0; c < C; ++c) {
        s += row[c] * a_src[h * C + c];
        d += row[c] * a_dst[h * C + c];
    }
    as[idx] = s;
    ad[idx] = d;
}

// ---------- edge pass 1: segment max (softmax stabilization) ----------
template <int H>
__global__ void edge_max(const int* __restrict__ srcs, const int* __restrict__ dsts,
                         int E, int N, const float* __restrict__ as,
                         const float* __restrict__ ad, unsigned* __restrict__ m) {
    int e = blockIdx.x * blockDim.x + threadIdx.x;
    if (e >= E + N) return;
    int s, d;
    if (e < E) { s = srcs[e]; d = dsts[e]; } else { s = d = e - E; }  // self-loops
#pragma unroll
    for (int h = 0; h < H; ++h) {
        float v = lrelu(as[s * H + h] + ad[d * H + h]);
        atomicMax(&m[d * H + h], fenc(v));
    }
}

// ---------- edge pass 2: exp-sum + unnormalized message aggregation ----------
template <int H, int C>
__global__ void edge_agg(const int* __restrict__ srcs, const int* __restrict__ dsts,
                         int E, int N, const float* __restrict__ as,
                         const float* __restrict__ ad, const unsigned* __restrict__ m,
                         const float* __restrict__ xl, float* __restrict__ ssum,
                         float* __restrict__ agg) {
    int e = blockIdx.x * blockDim.x + threadIdx.x;
    if (e >= E + N) return;
    int s, d;
    if (e < E) { s = srcs[e]; d = dsts[e]; } else { s = d = e - E; }
#pragma unroll
    for (int h = 0; h < H; ++h) {
        float v = lrelu(as[s * H + h] + ad[d * H + h]);
        float w = __expf(v - fdec(m[d * H + h]));
        atomicAdd(&ssum[d * H + h], w);
        const float* xr = xl  + (size_t)s * H * C + (size_t)h * C;
        float*       ar = agg + (size_t)d * H * C + (size_t)h * C;
#pragma unroll
        for (int c = 0; c < C; ++c) atomicAdd(&ar[c], w * xr[c]);
    }
}

// ---------- layer-1 finalize: normalize + bias + ELU (in place) ----------
__global__ void finalize1(float* __restrict__ agg, const float* __restrict__ ssum,
                          const float* __restrict__ b, int N) {
    long idx = (long)blockIdx.x * blockDim.x + threadIdx.x;
    if (idx >= (long)N * 64) return;
    int n = (int)(idx >> 6), c = (int)(idx & 63);
    int h = c >> 5;
    float v = agg[idx] / (ssum[n * 2 + h] + 1e-16f) + b[c];
    agg[idx] = elu(v);
}

// ---------- layer-2 finalize + output head: normalize + bias + ELU + @Wo + sigmoid ----------
__global__ void finalize2(const float* __restrict__ agg2, const float* __restrict__ s2,
                          const float* __restrict__ b2, const float* __restrict__ Wo,
                          const float* __restrict__ bo, float* __restrict__ out, int N) {
    int n = blockIdx.x * blockDim.x + threadIdx.x;
    if (n >= N) return;
    float inv = 1.f / (s2[n] + 1e-16f);
    float dot = 0.f;
#pragma unroll
    for (int c = 0; c < 16; ++c) {
        float v = elu(agg2[(size_t)n * 16 + c] * inv + b2[c]);
        dot += v * Wo[c];
    }
    out[n] = 1.f / (1.f + __expf(-(dot + bo[0])));
}

extern "C" void kernel_launch(void* const* d_in, const int* in_sizes, int n_in,
                              void* d_out, int out_size, void* d_ws, size_t ws_size,
                              hipStream_t stream) {
    const float* x      = (const float*)d_in[0];
    const int*   ei     = (const int*)d_in[1];
    const float* W1     = (const float*)d_in[2];
    const float* a_src1 = (const float*)d_in[3];
    const float* a_dst1 = (const float*)d_in[4];
    const float* b1     = (const float*)d_in[5];
    const float* W2     = (const float*)d_in[6];
    const float* a_src2 = (const float*)d_in[7];
    const float* a_dst2 = (const float*)d_in[8];
    const float* b2     = (const float*)d_in[9];
    const float* Wo     = (const float*)d_in[10];
    const float* bo     = (const float*)d_in[11];
    float* out = (float*)d_out;

    const int N = in_sizes[0] / 64;   // 100000
    const int E = in_sizes[1] / 2;    // 1600000
    const int ET = E + N;             // with self-loops
    const int* src = ei;
    const int* dst = ei + E;

    // workspace carve-up (floats)
    float* ws = (float*)d_ws;
    size_t off = 0;
    float*    xl1  = ws + off; off += (size_t)N * 64;
    float*    as1  = ws + off; off += (size_t)N * 2;
    float*    ad1  = ws + off; off += (size_t)N * 2;
    unsigned* m1   = (unsigned*)(ws + off); off += (size_t)N * 2;
    float*    s1   = ws + off; off += (size_t)N * 2;
    float*    agg1 = ws + off; off += (size_t)N * 64;   // becomes h1 in place
    float*    xl2  = ws + off; off += (size_t)N * 16;
    float*    as2  = ws + off; off += (size_t)N;
    float*    ad2  = ws + off; off += (size_t)N;
    unsigned* m2   = (unsigned*)(ws + off); off += (size_t)N;
    float*    s2   = ws + off; off += (size_t)N;
    float*    agg2 = ws + off; off += (size_t)N * 16;

    const unsigned NEG_INF_ENC = 0x007FFFFFu;  // fenc(-inf)
    const int TB = 256;
    auto blocks = [](long n, int tb) { return (unsigned)((n + tb - 1) / tb); };

    // ===== Layer 1 =====
    {
        dim3 g((N + 15) / 16, 64 / 16);
        wmma_gemm_f32<<<g, 32, 0, stream>>>(x, W1, xl1, N, 64, 64);
    }
    alpha_kernel<2, 32><<<blocks((long)N * 2, TB), TB, 0, stream>>>(xl1, a_src1, a_dst1, as1, ad1, N);
    fill_u32<<<blocks((long)N * 2, TB), TB, 0, stream>>>(m1, NEG_INF_ENC, (long)N * 2);
    fill_f32<<<blocks((long)N * 2, TB), TB, 0, stream>>>(s1, 0.f, (long)N * 2);
    fill_f32<<<blocks((long)N * 64, TB), TB, 0, stream>>>(agg1, 0.f, (long)N * 64);
    edge_max<2><<<blocks(ET, TB), TB, 0, stream>>>(src, dst, E, N, as1, ad1, m1);
    edge_agg<2, 32><<<blocks(ET, TB), TB, 0, stream>>>(src, dst, E, N, as1, ad1, m1, xl1, s1, agg1);
    finalize1<<<blocks((long)N * 64, TB), TB, 0, stream>>>(agg1, s1, b1, N);

    // ===== Layer 2 =====
    {
        dim3 g((N + 15) / 16, 1);
        wmma_gemm_f32<<<g, 32, 0, stream>>>(agg1, W2, xl2, N, 64, 16);
    }
    alpha_kernel<1, 16><<<blocks(N, TB), TB, 0, stream>>>(xl2, a_src2, a_dst2, as2, ad2, N);
    fill_u32<<<blocks(N, TB), TB, 0, stream>>>(m2, NEG_INF_ENC, N);
    fill_f32<<<blocks(N, TB), TB, 0, stream>>>(s2, 0.f, N);
    fill_f32<<<blocks((long)N * 16, TB), TB, 0, stream>>>(agg2, 0.f, (long)N * 16);
    edge_max<1><<<blocks(ET, TB), TB, 0, stream>>>(src, dst, E, N, as2, ad2, m2);
    edge_agg<1, 16><<<blocks(ET, TB), TB, 0, stream>>>(src, dst, E, N, as2, ad2, m2, xl2, s2, agg2);

    // ===== Output head =====
    finalize2<<<blocks(N, TB), TB, 0, stream>>>(agg2, s2, b2, Wo, bo, out, N);
}